// GATModel_19155554140770
// MI455X (gfx1250) — compile-verified
//
#include <hip/hip_runtime.h>

#define CN0 512
#define CN1 2048
#define CN2 8192
#define KNB 33
#define CIN 256
#define CHID 128
#define COUT 128
#define CNH 8
#define LRELU_ALPHA 0.2f

typedef float v2f __attribute__((ext_vector_type(2)));
typedef float v8f __attribute__((ext_vector_type(8)));

// ---- pack W1 [NH, IN, HID] -> Bp1: K-pair interleaved [IN/2][NH*HID][2] ----
// Bp[(k>>1)*2N + 2n + (k&1)] = B[k,n]  with  n = h*HID + j, k = i
__global__ void gat_pack_w1_kernel(const float* __restrict__ W1,
                                   float* __restrict__ Bp1) {
  int idx = blockIdx.x * blockDim.x + threadIdx.x;   // < NH*IN*HID
  int j = idx % CHID;
  int i = (idx / CHID) % CIN;
  int h = idx / (CHID * CIN);
  int n = h * CHID + j;
  Bp1[(size_t)(i >> 1) * (2 * CNH * CHID) + 2 * n + (i & 1)] = W1[idx];
}

// ---- pack W2 [NH*HID, OUT] (row-major) into same K-pair layout ----
__global__ void gat_pack_w2_kernel(const float* __restrict__ W2,
                                   float* __restrict__ Bp2) {
  int idx = blockIdx.x * blockDim.x + threadIdx.x;   // < NH*HID*OUT
  int o = idx % COUT;
  int c = idx / COUT;
  Bp2[(size_t)(c >> 1) * (2 * COUT) + 2 * o + (c & 1)] = W2[idx];
}

// ---- fp32 WMMA GEMM, B pre-packed in K-pair layout ----
// one wave computes a 32x64 C tile; grid = (N/64, M/32), block = 32
__global__ void gat_wmma_gemm_f32_kernel(const float* __restrict__ A,
                                         const float* __restrict__ Bp,
                                         float* __restrict__ C,
                                         int K, int N, int lda, int ldc) {
  const int lane = threadIdx.x & 31;
  const int hi   = lane >> 4;      // 0/1 half-wave
  const int lr   = lane & 15;      // M (A) / N (B,C) within 16-tile
  const int m0   = blockIdx.y << 5;
  const int n0   = blockIdx.x << 6;

  v8f acc0 = {}; v8f acc1 = {}; v8f acc2 = {}; v8f acc3 = {};
  v8f acc4 = {}; v8f acc5 = {}; v8f acc6 = {}; v8f acc7 = {};

  const float* Arow0 = A + (size_t)(m0 + lr) * lda;       // rows m0..m0+15
  const float* Arow1 = Arow0 + (size_t)16 * lda;          // rows m0+16..m0+31
  const v2f* Bv = (const v2f*)Bp;                         // one K-pair per elem

  // software-pipelined operand registers
  v2f a0, a1, b0, b1, b2, b3;
  {
    const int ka = 2 * hi;                                // k = 0
    a0 = *(const v2f*)(Arow0 + ka);
    a1 = *(const v2f*)(Arow1 + ka);
    const v2f* bp = Bv + (size_t)hi * N + n0 + lr;        // (ka>>1)*N
    b0 = bp[0]; b1 = bp[16]; b2 = bp[32]; b3 = bp[48];
  }
  for (int k = 0; k < K; k += 4) {
    const v2f ca0 = a0, ca1 = a1;
    const v2f cb0 = b0, cb1 = b1, cb2 = b2, cb3 = b3;
    if (k + 4 < K) {                                      // prefetch next k-step
      const int ka = k + 4 + 2 * hi;
      a0 = *(const v2f*)(Arow0 + ka);
      a1 = *(const v2f*)(Arow1 + ka);
      const v2f* bp = Bv + (size_t)(ka >> 1) * N + n0 + lr;
      b0 = bp[0]; b1 = bp[16]; b2 = bp[32]; b3 = bp[48];
    }
    acc0 = __builtin_amdgcn_wmma_f32_16x16x4_f32(false, ca0, false, cb0, (short)0, acc0, false, false);
    acc1 = __builtin_amdgcn_wmma_f32_16x16x4_f32(false, ca0, false, cb1, (short)0, acc1, false, false);
    acc2 = __builtin_amdgcn_wmma_f32_16x16x4_f32(false, ca0, false, cb2, (short)0, acc2, false, false);
    acc3 = __builtin_amdgcn_wmma_f32_16x16x4_f32(false, ca0, false, cb3, (short)0, acc3, false, false);
    acc4 = __builtin_amdgcn_wmma_f32_16x16x4_f32(false, ca1, false, cb0, (short)0, acc4, false, false);
    acc5 = __builtin_amdgcn_wmma_f32_16x16x4_f32(false, ca1, false, cb1, (short)0, acc5, false, false);
    acc6 = __builtin_amdgcn_wmma_f32_16x16x4_f32(false, ca1, false, cb2, (short)0, acc6, false, false);
    acc7 = __builtin_amdgcn_wmma_f32_16x16x4_f32(false, ca1, false, cb3, (short)0, acc7, false, false);
  }

  // C/D layout: VGPR i holds (M = i + 8*hi, N = lr) within each 16x16 tile
  float* Crow0 = C + (size_t)(m0 + 8 * hi) * ldc + n0 + lr;
  float* Crow1 = Crow0 + (size_t)16 * ldc;
#pragma unroll
  for (int i = 0; i < 8; ++i) {
    Crow0[(size_t)i * ldc + 0]  = acc0[i];
    Crow0[(size_t)i * ldc + 16] = acc1[i];
    Crow0[(size_t)i * ldc + 32] = acc2[i];
    Crow0[(size_t)i * ldc + 48] = acc3[i];
    Crow1[(size_t)i * ldc + 0]  = acc4[i];
    Crow1[(size_t)i * ldc + 16] = acc5[i];
    Crow1[(size_t)i * ldc + 32] = acc6[i];
    Crow1[(size_t)i * ldc + 48] = acc7[i];
  }
}

// ---- per-(node,head) attention dots: S[w*2+0]=h·a[:128], +1 = h·a[128:] ----
__global__ void gat_dot_kernel(const float* __restrict__ Hm,
                               const float* __restrict__ a,
                               float* __restrict__ S,
                               int nwaves, int nheads, int rowStride) {
  int gt   = blockIdx.x * blockDim.x + threadIdx.x;
  int w    = gt >> 5;
  int lane = gt & 31;
  if (w >= nwaves) return;
  int n = w / nheads, h = w % nheads;
  const float* hp = Hm + (size_t)n * rowStride + h * CHID;
  const float* ap = a + (size_t)h * 2 * CHID;
  float sA = 0.f, sB = 0.f;
#pragma unroll
  for (int j = lane; j < CHID; j += 32) {
    float v = hp[j];
    sA += v * ap[j];
    sB += v * ap[CHID + j];
  }
#pragma unroll
  for (int off = 16; off > 0; off >>= 1) {
    sA += __shfl_down(sA, off, 32);
    sB += __shfl_down(sB, off, 32);
  }
  if (lane == 0) { S[2 * w] = sA; S[2 * w + 1] = sB; }
}

// ---- attention softmax + gather-weighted sum + ELU ----
__global__ void gat_agg_kernel(const float* __restrict__ Hm,
                               const float* __restrict__ S,
                               const int* __restrict__ src,
                               const int* __restrict__ dst,
                               float* __restrict__ Yout,
                               int nheads, int rowStride) {
  const int b = blockIdx.x;
  const int t = b / nheads;
  const int h = b % nheads;
  __shared__ float wsh[KNB];
  __shared__ int   dsh[KNB];
  const int tid = threadIdx.x;

  if (tid < KNB) {
    int m = t * KNB + tid;
    int s = src[m], d = dst[m];
    float e = S[(size_t)(s * nheads + h) * 2] + S[(size_t)(d * nheads + h) * 2 + 1];
    wsh[tid] = (e > 0.f) ? e : LRELU_ALPHA * e;   // LeakyReLU
    dsh[tid] = d;
  }
  __syncthreads();
  if (tid == 0) {        // K=33: serial softmax is negligible
    float mx = wsh[0];
    for (int k = 1; k < KNB; ++k) mx = fmaxf(mx, wsh[k]);
    float sum = 0.f;
    for (int k = 0; k < KNB; ++k) { float ex = expf(wsh[k] - mx); wsh[k] = ex; sum += ex; }
    float inv = 1.f / sum;
    for (int k = 0; k < KNB; ++k) wsh[k] *= inv;
  }
  __syncthreads();

  float acc = 0.f;
  const float* col = Hm + (size_t)h * CHID + tid;
#pragma unroll 4
  for (int k = 0; k < KNB; ++k)
    acc += wsh[k] * col[(size_t)dsh[k] * rowStride];
  acc = (acc > 0.f) ? acc : expm1f(acc);          // ELU
  Yout[(size_t)t * (nheads * CHID) + h * CHID + tid] = acc;
}

extern "C" void kernel_launch(void* const* d_in, const int* in_sizes, int n_in,
                              void* d_out, int out_size, void* d_ws, size_t ws_size,
                              hipStream_t stream) {
  (void)in_sizes; (void)n_in; (void)out_size; (void)ws_size;
  const float* feats = (const float*)d_in[0];
  const float* W1    = (const float*)d_in[1];
  const float* a1    = (const float*)d_in[2];
  const float* W2    = (const float*)d_in[3];
  const float* a2    = (const float*)d_in[4];
  const int*   src1  = (const int*)d_in[5];
  const int*   dst1  = (const int*)d_in[6];
  const int*   src2  = (const int*)d_in[8];
  const int*   dst2  = (const int*)d_in[9];
  float* out = (float*)d_out;

  // workspace layout (floats); every region fully written before read
  float* ws  = (float*)d_ws;
  float* h1  = ws;                                   // [8192, 1024]
  float* x   = h1  + (size_t)CN2 * CNH * CHID;       // [2048, 1024]
  float* Bp1 = x   + (size_t)CN1 * CNH * CHID;       // [256*1024] packed W1
  float* Bp2 = Bp1 + (size_t)CIN * CNH * CHID;       // [1024*128] packed W2
  float* h2  = Bp2 + (size_t)CNH * CHID * COUT;      // [2048, 128]
  float* s1  = h2  + (size_t)CN1 * COUT;             // [8192*8, 2]
  float* s2  = s1  + (size_t)CN2 * CNH * 2;          // [2048, 2]

  // layer 1
  gat_pack_w1_kernel<<<(CNH * CIN * CHID) / 256, 256, 0, stream>>>(W1, Bp1);
  gat_pack_w2_kernel<<<(CNH * CHID * COUT) / 256, 256, 0, stream>>>(W2, Bp2);
  dim3 g1((CNH * CHID) / 64, CN2 / 32);
  gat_wmma_gemm_f32_kernel<<<g1, 32, 0, stream>>>(feats, Bp1, h1,
                                                  CIN, CNH * CHID, CIN, CNH * CHID);
  gat_dot_kernel<<<(CN2 * CNH) / 8, 256, 0, stream>>>(h1, a1, s1,
                                                      CN2 * CNH, CNH, CNH * CHID);
  gat_agg_kernel<<<CN1 * CNH, 128, 0, stream>>>(h1, s1, src1, dst1, x,
                                                CNH, CNH * CHID);
  // layer 2
  dim3 g2(COUT / 64, CN1 / 32);
  gat_wmma_gemm_f32_kernel<<<g2, 32, 0, stream>>>(x, Bp2, h2,
                                                  CNH * CHID, COUT, CNH * CHID, COUT);
  gat_dot_kernel<<<CN1 / 8, 256, 0, stream>>>(h2, a2, s2, CN1, 1, COUT);
  gat_agg_kernel<<<CN0, 128, 0, stream>>>(h2, s2, src2, dst2, out, 1, COUT);
}